// DynamicGroup_65377992180033
// MI455X (gfx1250) — compile-verified
//
#include <hip/hip_runtime.h>
#include <cstdint>

typedef float v2f __attribute__((ext_vector_type(2)));
typedef float v4f __attribute__((ext_vector_type(4)));
typedef float v8f __attribute__((ext_vector_type(8)));

#define B_DIM 64
#define H_DIM 1024
#define I_DIM 512

// ---------------------------------------------------------------------------
// Kernel A: input_contrib = x_t @ W_ih^T + b_ih  via V_WMMA_F32_16X16X4_F32
// One wave per 16x16 tile of the (64 x 1024) output. 4 waves per block.
// ---------------------------------------------------------------------------
__global__ __launch_bounds__(128) void reservoir_gemm_wmma(
    const float* __restrict__ x_t,   // (64, 512) row-major
    const float* __restrict__ W_ih,  // (1024, 512) row-major
    const float* __restrict__ b_ih,  // (1024)
    float* __restrict__ ic)          // (64, 1024) output
{
  const int lane = threadIdx.x & 31;
  const int waveInBlk = threadIdx.x >> 5;
  const int tile = blockIdx.x * 4 + waveInBlk;  // 256 tiles total
  const int tileM = tile & 3;                   // 64/16 = 4 tiles along B
  const int tileN = tile >> 2;                  // 1024/16 = 64 tiles along H

  const int mLane = lane & 15;
  const int kBase = (lane >> 4) << 1;  // lanes 0-15 hold K,K+1 of {0,1}; 16-31 of {2,3}

  // A-frag: row (tileM*16 + mLane) of x_t, two consecutive K per lane.
  const float* aPtr = x_t + (size_t)(tileM * 16 + mLane) * I_DIM + kBase;
  // B-frag: B[k][n] = W_ih[n][k] -> row (tileN*16 + mLane) of W_ih, same K pair.
  const float* bPtr = W_ih + (size_t)(tileN * 16 + mLane) * I_DIM + kBase;

  v8f acc = {};
  for (int k0 = 0; k0 < I_DIM; k0 += 4) {
    v2f a = *reinterpret_cast<const v2f*>(aPtr + k0);
    v2f b = *reinterpret_cast<const v2f*>(bPtr + k0);
    acc = __builtin_amdgcn_wmma_f32_16x16x4_f32(
        /*neg_a=*/false, a, /*neg_b=*/false, b,
        /*c_mod=*/(short)0, acc, /*reuse_a=*/false, /*reuse_b=*/false);
  }

  // C layout: VGPR r -> M = r (lanes 0-15) or r+8 (lanes 16-31), N = lane&15.
  const int col = tileN * 16 + mLane;
  const float bias = b_ih[col];
  const int rowBase = tileM * 16 + ((lane >> 4) << 3);
#pragma unroll
  for (int r = 0; r < 8; ++r) {
    ic[(size_t)(rowBase + r) * H_DIM + col] = acc[r] + bias;
  }
}

// ---------------------------------------------------------------------------
// Kernel B: fused  h_next = tanh(ic + softmax(W_hh/tau + g) . h_prev)
// Block = 256 threads = 8 waves; one batch b, 8 rows o per block.
// Each wave holds its full 1024-element logit row in registers (32 lanes x
// 8 float4), so the exact two-pass softmax costs zero extra HBM traffic.
// Gumbel stream (256 MB, touched once) loaded non-temporally to keep the
// L2-resident W_hh (4 MB, reused 64x) hot.
// ---------------------------------------------------------------------------
__global__ __launch_bounds__(256) void reservoir_softmax_mix(
    const float* __restrict__ h_prev,       // (64, 1024)
    const float* __restrict__ W_hh,         // (1024, 1024)
    const float* __restrict__ temperature,  // scalar
    const float* __restrict__ gumbel,       // (64, 1024, 1024)
    const float* __restrict__ ic,           // (64, 1024) from kernel A
    float* __restrict__ out)                // (64, 1024)
{
  __shared__ float hp[H_DIM];

  const int b = blockIdx.x >> 7;               // 128 blocks per batch
  const int oBase = (blockIdx.x & 127) << 3;   // 8 rows per block
  const int tid = threadIdx.x;

  // Stage h_prev[b, :] into LDS (256 threads x float4 = 1024 floats).
  {
    const v4f* src = reinterpret_cast<const v4f*>(h_prev + (size_t)b * H_DIM);
    reinterpret_cast<v4f*>(hp)[tid] = src[tid];
  }
  __syncthreads();

  const float tau = fmaxf(temperature[0], 1e-3f);
  const float inv_tau = 1.0f / tau;

  const int lane = tid & 31;
  const int wv = tid >> 5;
  const int o = oBase + wv;

  const float* grow = gumbel + ((size_t)(b * H_DIM + o)) * H_DIM;
  const float* wrow = W_hh + (size_t)o * H_DIM;

  // Pass 1: build logits in registers, track running max.
  v4f v[8];
  float m = -3.0e38f;
#pragma unroll
  for (int it = 0; it < 8; ++it) {
    const int h0 = it * 128 + lane * 4;
    v4f g = __builtin_nontemporal_load(reinterpret_cast<const v4f*>(grow + h0));
    v4f w = *reinterpret_cast<const v4f*>(wrow + h0);
#pragma unroll
    for (int c = 0; c < 4; ++c) {
      float t = fmaf(w[c], inv_tau, g[c]);
      v[it][c] = t;
      m = fmaxf(m, t);
    }
  }
  // Wave32 max-reduction.
#pragma unroll
  for (int off = 16; off > 0; off >>= 1)
    m = fmaxf(m, __shfl_xor(m, off, 32));

  // Pass 2: sum of exp and exp-weighted h_prev, all from registers/LDS.
  float s = 0.0f, wsum = 0.0f;
#pragma unroll
  for (int it = 0; it < 8; ++it) {
    const int h0 = it * 128 + lane * 4;
    v4f hpv = *reinterpret_cast<const v4f*>(&hp[h0]);
#pragma unroll
    for (int c = 0; c < 4; ++c) {
      float e = __expf(v[it][c] - m);
      s += e;
      wsum = fmaf(e, hpv[c], wsum);
    }
  }
#pragma unroll
  for (int off = 16; off > 0; off >>= 1) {
    s += __shfl_xor(s, off, 32);
    wsum += __shfl_xor(wsum, off, 32);
  }

  if (lane == 0) {
    const float contrib = wsum / s;
    out[(size_t)b * H_DIM + o] = tanhf(ic[(size_t)b * H_DIM + o] + contrib);
  }
}

// ---------------------------------------------------------------------------
extern "C" void kernel_launch(void* const* d_in, const int* in_sizes, int n_in,
                              void* d_out, int out_size, void* d_ws, size_t ws_size,
                              hipStream_t stream) {
  const float* x_t         = (const float*)d_in[0];
  const float* h_prev      = (const float*)d_in[1];
  const float* W_ih        = (const float*)d_in[2];
  const float* b_ih        = (const float*)d_in[3];
  const float* W_hh        = (const float*)d_in[4];
  const float* temperature = (const float*)d_in[5];
  const float* gumbel      = (const float*)d_in[6];
  float* out = (float*)d_out;
  float* ic  = (float*)d_ws;  // 64*1024 floats = 256 KB scratch

  // Kernel A: 256 tiles, 4 waves/block -> 64 blocks of 128 threads.
  reservoir_gemm_wmma<<<64, 128, 0, stream>>>(x_t, W_ih, b_ih, ic);

  // Kernel B: 64 batches * 128 row-groups = 8192 blocks of 256 threads.
  reservoir_softmax_mix<<<B_DIM * (H_DIM / 8), 256, 0, stream>>>(
      h_prev, W_hh, temperature, gumbel, ic, out);
}